// SC_61254823576197
// MI455X (gfx1250) — compile-verified
//
#include <hip/hip_runtime.h>
#include <stdint.h>

typedef float v2f  __attribute__((ext_vector_type(2)));
typedef float v8f  __attribute__((ext_vector_type(8)));
typedef unsigned int v4u __attribute__((ext_vector_type(4)));
typedef int v8i __attribute__((ext_vector_type(8)));
typedef int v4i __attribute__((ext_vector_type(4)));

#define N_ROWS  4096
#define IN_DIM  64
#define OUT_DIM 64
#define TAPS    4
#define KDIM    (N_ROWS * TAPS)   // 16384 flattened reduction length
#define NSEG    8                 // K-split for parallelism
#define SEGLEN  (KDIM / NSEG)     // 2048
#define KC      256               // chunk width in floats (must stay 256: pad_interval code 7)
#define CHUNKS  (SEGLEN / KC)     // 8
#define ROWS    64                // A rows per workgroup
#define LDS_PAD 4
#define LDS_STRIDE (KC + LDS_PAD) // 260 floats/row -> all-64-bank WMMA-A fragment reads

static_assert(KC == 256, "TDM pad_interval encoding assumes 256-dword rows");

// ---------------------------------------------------------------------------
// Tensor Data Mover: DMA a ROWS x KC fp32 tile (row stride KDIM elements in
// global) into LDS, padding each 256-dword row with 4 dwords (stride 260).
// Descriptor layout per CDNA5 ISA 8.3/8.4 (D# group0 128b, group1 256b).
// Toolchain here is the 6-arg builtin variant (clang-23 / therock-10.0).
// ---------------------------------------------------------------------------
__device__ __forceinline__ void SC_tdm_load(const float* gsrc, unsigned lds_byte_off) {
  unsigned long long ga = (unsigned long long)(uintptr_t)gsrc;
  v4u g0;
  g0.x = 1u;                                        // count=1, user descriptor
  g0.y = lds_byte_off;                              // lds_addr
  g0.z = (unsigned)(ga & 0xFFFFFFFFull);            // global_addr[31:0]
  g0.w = (unsigned)((ga >> 32) & 0x1FFFFFFull)      // global_addr[56:32]
       | (2u << 30);                                // type = 2 ("image")
  v8i g1;
  g1[0] = (int)((2u << 16)                          // data_size = 4 bytes
              | (1u << 20)                          // pad_enable
              | (7u << 22)                          // pad_interval = 256 dwords
              | (3u << 25));                        // pad_amount   = 4 dwords
  g1[1] = (int)(((unsigned)KDIM   & 0xFFFFu) << 16);// tensor_dim0 lo16 (=16384)
  g1[2] = (int)(((unsigned)N_ROWS & 0xFFFFu) << 16);// tensor_dim1 lo16 (=4096)
  g1[3] = (int)(((unsigned)KC) << 16);              // tile_dim0 = 256
  g1[4] = (int)ROWS;                                // tile_dim1 = 64, tile_dim2 = 0
  g1[5] = (int)KDIM;                                // tensor_dim0_stride = 16384
  g1[6] = 0;
  g1[7] = 0;
  v4i z4 = {0, 0, 0, 0};
  v8i z8 = {0, 0, 0, 0, 0, 0, 0, 0};
  __builtin_amdgcn_tensor_load_to_lds(g0, g1, z4, z4, z8, 0);
}

// ---------------------------------------------------------------------------
// Stage 1: BP[(m*4+k)*64+o] = sum_i x[m,i]*P[i,o,k]  (and BQ from Q).
// coeff layout [IN,OUT,K] row-major => P[i*256 + o*4 + k] = P[i*256 + tid].
// ---------------------------------------------------------------------------
__global__ void __launch_bounds__(256)
SC_61254823576197_proj(const float* __restrict__ x,
                       const float* __restrict__ P,
                       const float* __restrict__ Q,
                       float* __restrict__ BP, float* __restrict__ BQ) {
  __shared__ float xs[IN_DIM];
  const int m = blockIdx.x;
  const int t = threadIdx.x;            // t = o*4 + k
  if (t < IN_DIM) xs[t] = x[m * IN_DIM + t];
  __syncthreads();
  const int o = t >> 2, k = t & 3;
  float sp = 0.f, sq = 0.f;
  #pragma unroll 8
  for (int i = 0; i < IN_DIM; ++i) {
    const float xv = xs[i];
    sp += xv * P[i * (OUT_DIM * TAPS) + t];
    sq += xv * Q[i * (OUT_DIM * TAPS) + t];
  }
  BP[(m * TAPS + k) * OUT_DIM + o] = sp;
  BQ[(m * TAPS + k) * OUT_DIM + o] = sq;
}

// ---------------------------------------------------------------------------
// Stage 2: per (mblock, seg): partial[seg] += A[64 x 2048] @ B[2048 x 64]
// for both stacks, using V_WMMA_F32_16X16X4_F32. 4 waves, each owns one
// 16-col N-tile and 4 stacked 16x16 M-tiles. TDM double-buffers A into LDS.
// ---------------------------------------------------------------------------
__global__ void __launch_bounds__(128)
SC_61254823576197_gemm(const float* __restrict__ Ld, const float* __restrict__ Lu,
                       const float* __restrict__ BP, const float* __restrict__ BQ,
                       float* __restrict__ partial) {
  __shared__ float Abuf[2][ROWS * LDS_STRIDE];   // 2 x 66560 B
  const int mblock = blockIdx.x;                 // 0..63
  const int seg    = blockIdx.y;                 // 0..7
  const int tid  = threadIdx.x;
  const int wave = tid >> 5;                     // 0..3 -> N-tile
  const int lane = tid & 31;
  const int lo   = lane & 15;
  const int hi   = lane >> 4;                    // 0/1: K-pair select in A/B frags

  const unsigned lds_off0 = (unsigned)(uintptr_t)(&Abuf[0][0]);
  const unsigned lds_off1 = (unsigned)(uintptr_t)(&Abuf[1][0]);

  v8f acc[4];
  #pragma unroll
  for (int i = 0; i < 4; ++i)
    acc[i] = (v8f){0.f, 0.f, 0.f, 0.f, 0.f, 0.f, 0.f, 0.f};

  #pragma unroll 1
  for (int s = 0; s < 2; ++s) {
    const float* A = s ? Lu : Ld;
    const float* B = s ? BQ : BP;
    const float* Aseg = A + (size_t)mblock * ROWS * KDIM + (size_t)seg * SEGLEN;

    if (wave == 0) {
      SC_tdm_load(Aseg, lds_off0);
      __builtin_amdgcn_s_wait_tensorcnt(0);
    }
    __syncthreads();

    #pragma unroll 1
    for (int c = 0; c < CHUNKS; ++c) {
      const int cur = c & 1;
      if (wave == 0 && (c + 1) < CHUNKS)
        SC_tdm_load(Aseg + (size_t)(c + 1) * KC, cur ? lds_off0 : lds_off1);

      // B fragment base: VGPR0 <- row (kglob + hi*2), VGPR1 <- next row.
      const float* bp = B + (size_t)(seg * SEGLEN + c * KC + hi * 2) * OUT_DIM
                          + wave * 16 + lo;
      // A fragment base in LDS: lane lo = row, hi picks K{0,1}/K{2,3}.
      const float* lb = &Abuf[cur][0] + (size_t)lo * LDS_STRIDE + hi * 2;

      #pragma unroll 4
      for (int kb = 0; kb < KC; kb += 4) {
        v2f b;
        b.x = bp[0];
        b.y = bp[OUT_DIM];
        #pragma unroll
        for (int mt = 0; mt < 4; ++mt) {
          v2f a = *(const v2f*)(lb + mt * 16 * LDS_STRIDE + kb);  // ds_load_b64
          acc[mt] = __builtin_amdgcn_wmma_f32_16x16x4_f32(
              false, a, false, b, (short)0, acc[mt], false, false);
        }
        bp += 4 * OUT_DIM;
      }

      if (wave == 0) __builtin_amdgcn_s_wait_tensorcnt(0);
      __syncthreads();
    }
  }

  // C/D layout: VGPR j, lanes 0-15 -> M=j, lanes 16-31 -> M=8+j; N = lane&15.
  float* pout = partial + (size_t)seg * (N_ROWS * OUT_DIM)
                        + (size_t)(mblock * ROWS) * OUT_DIM
                        + (size_t)(wave * 16 + lo);
  #pragma unroll
  for (int mt = 0; mt < 4; ++mt) {
    #pragma unroll
    for (int j = 0; j < 8; ++j)
      pout[(size_t)(mt * 16 + hi * 8 + j) * OUT_DIM] = acc[mt][j];
  }
}

// ---------------------------------------------------------------------------
// Stage 3: out = sum over the 8 K-segment partials (deterministic reduce).
// ---------------------------------------------------------------------------
__global__ void __launch_bounds__(256)
SC_61254823576197_reduce(const float* __restrict__ partial, float* __restrict__ out) {
  const int i = blockIdx.x * 256 + threadIdx.x;   // 0..262143
  float sum = 0.f;
  #pragma unroll
  for (int s = 0; s < NSEG; ++s)
    sum += partial[(size_t)s * (N_ROWS * OUT_DIM) + i];
  out[i] = sum;
}

extern "C" void kernel_launch(void* const* d_in, const int* in_sizes, int n_in,
                              void* d_out, int out_size, void* d_ws, size_t ws_size,
                              hipStream_t stream) {
  const float* Ld = (const float*)d_in[0];   // [4096,4096,4]
  const float* Lu = (const float*)d_in[1];   // [4096,4096,4]
  const float* x  = (const float*)d_in[2];   // [4096,64]
  const float* P  = (const float*)d_in[3];   // [64,64,4]
  const float* Q  = (const float*)d_in[4];   // [64,64,4]

  float* ws = (float*)d_ws;
  float* BP      = ws;                       // 16384*64 floats (4 MB)
  float* BQ      = ws + (size_t)KDIM * OUT_DIM;
  float* partial = ws + (size_t)2 * KDIM * OUT_DIM;  // 8 * 4096*64 floats (8 MB)

  SC_61254823576197_proj<<<N_ROWS, 256, 0, stream>>>(x, P, Q, BP, BQ);
  SC_61254823576197_gemm<<<dim3(N_ROWS / ROWS, NSEG), 128, 0, stream>>>(
      Ld, Lu, BP, BQ, partial);
  SC_61254823576197_reduce<<<(N_ROWS * OUT_DIM) / 256, 256, 0, stream>>>(
      partial, (float*)d_out);
}